// TailSTEAK_56538949484989
// MI455X (gfx1250) — compile-verified
//
#include <hip/hip_runtime.h>
#include <hip/hip_bf16.h>

// ---------------------------------------------------------------------------
// GCN (2-layer) + link prediction head for MI455X (gfx1250, wave32, WMMA).
// fp32 throughout (matches reference numerics); GEMMs use V_WMMA_F32_16X16X4_F32.
// LDS strides padded for the 64-bank LDS: A stride 132, W stride 144 -> the
// b64 A loads hit all 64 banks once; B load half-wave windows are disjoint.
// Persistent GEMM blocks keep W resident in LDS across M tiles.
// ---------------------------------------------------------------------------

typedef __attribute__((ext_vector_type(2))) float v2f;
typedef __attribute__((ext_vector_type(8))) float v8f;

#define NNODES 50000
#define NEDGES 800000
#define DFEAT  128
#define NBATCH 2048
#define NNEG   19

#define ASTR 132   // padded LDS row stride (dwords) for A strip, 528B (16B aligned)
#define WSTR 144   // padded LDS row stride (dwords) for W,       576B (16B aligned)

// ------------------------- degree / dinv ----------------------------------
__global__ void deg_init_kernel(float* deg, int n) {
    int i = blockIdx.x * blockDim.x + threadIdx.x;
    if (i < n) deg[i] = 1.0f;               // self loop
}

__global__ void deg_edges_kernel(const int* __restrict__ dst, float* deg, int e) {
    int i = blockIdx.x * blockDim.x + threadIdx.x;
    if (i < e) atomicAdd(&deg[dst[i]], 1.0f);
}

__global__ void dinv_kernel(float* deg, int n) {    // in-place deg -> 1/sqrt(deg)
    int i = blockIdx.x * blockDim.x + threadIdx.x;
    if (i < n) deg[i] = rsqrtf(deg[i]);     // deg >= 1 always (self loop)
}

// ------------------------- GEMM: C = A(Mx128) * W(128x128) -----------------
// Persistent blocks: W staged once in LDS, blocks strip-mine 16-row M tiles.
// Block = 256 threads = 8 waves; wave w computes columns [16w, 16w+16).
__global__ void gemm_xw_kernel(const float* __restrict__ A,
                               const float* __restrict__ W,
                               float* __restrict__ C, int mtiles) {
    __shared__ float As[16 * ASTR];          //  8.25 KB
    __shared__ float Ws[DFEAT * WSTR];       // 72 KB

    const int tid  = threadIdx.x;
    const int wave = tid >> 5;
    const int lane = tid & 31;

    // Stage W once: 4096 float4 groups, 16 per thread, into padded rows.
    {
        const float4* Wv = (const float4*)W;
        #pragma unroll
        for (int g = tid; g < DFEAT * DFEAT / 4; g += 256) {
            int row = g >> 5, c4 = g & 31;
            *(float4*)&Ws[row * WSTR + c4 * 4] = Wv[g];
        }
    }

    const int ncol = wave * 16;        // this wave's output-column base
    const int half = lane >> 4;        // 0: lanes 0-15 (K+0,K+1); 1: lanes 16-31 (K+2,K+3)
    const int l15  = lane & 15;

    for (int mt = blockIdx.x; mt < mtiles; mt += gridDim.x) {
        const int mbase = mt * 16;
        __syncthreads();               // Ws staged / previous As readers done
        // Stage A strip: 512 float4 groups, 2 per thread, into padded rows.
        {
            const float4* Av = (const float4*)(A + (long)mbase * DFEAT);
            #pragma unroll
            for (int g = tid; g < 16 * DFEAT / 4; g += 256) {
                int row = g >> 5, c4 = g & 31;
                *(float4*)&As[row * ASTR + c4 * 4] = Av[g];
            }
        }
        __syncthreads();

        v8f acc = {};
        #pragma unroll 4
        for (int kk = 0; kk < DFEAT; kk += 4) {
            const int ka = kk + half * 2;
            v2f a, b;
            // A 16x4 fragment (ISA 7.12.2): lanes 0-15 row=l15 K={ka,ka+1}; 16-31 K={ka+2,ka+3}
            a.x = As[l15 * ASTR + ka];
            a.y = As[l15 * ASTR + ka + 1];
            // B 4x16 fragment: K rows striped across VGPRs, N across lanes
            b.x = Ws[ka * WSTR + ncol + l15];
            b.y = Ws[(ka + 1) * WSTR + ncol + l15];
            acc = __builtin_amdgcn_wmma_f32_16x16x4_f32(
                /*neg_a=*/false, a, /*neg_b=*/false, b,
                /*c_mod=*/(short)0, acc, /*reuse_a=*/false, /*reuse_b=*/false);
        }

        // C/D layout: VGPR v -> row v + 8*half, col = l15.
        // Single base pointer; row step is a constant 512B -> immediate offsets.
        float* Cp = C + ((long)(mbase + half * 8) * DFEAT + ncol + l15);
        #pragma unroll
        for (int v = 0; v < 8; ++v) {
            Cp[v * DFEAT] = acc[v];
        }
    }
}

// ------------------- aggregation: self-loop init + edge scatter ------------
// out[i] = h[i] * dinv[i]^2   (also serves as buffer initialization)
__global__ void agg_self_kernel(const float* __restrict__ h,
                                const float* __restrict__ dinv,
                                float* __restrict__ out, int n) {
    int idx = blockIdx.x * blockDim.x + threadIdx.x;   // over n * 32 float4 groups
    if (idx >= n * (DFEAT / 4)) return;
    int i = idx >> 5;
    float s = dinv[i] * dinv[i];
    float4 hv = ((const float4*)h)[idx];
    float4 o;
    o.x = hv.x * s; o.y = hv.y * s; o.z = hv.z * s; o.w = hv.w * s;
    ((float4*)out)[idx] = o;
}

// out[dst] += h[src] * dinv[src]*dinv[dst], one thread per (edge, 4 features)
__global__ void agg_edges_kernel(const float* __restrict__ h,
                                 const float* __restrict__ dinv,
                                 const int* __restrict__ src,
                                 const int* __restrict__ dst,
                                 float* __restrict__ out, int e) {
    int idx = blockIdx.x * blockDim.x + threadIdx.x;   // over e * 32
    if (idx >= e * (DFEAT / 4)) return;
    int ed = idx >> 5;
    int g  = idx & 31;
    int s  = src[ed], d = dst[ed];
    float norm = dinv[s] * dinv[d];
    float4 hv = ((const float4*)(h + (long)s * DFEAT))[g];
    float* o = out + (long)d * DFEAT + g * 4;
    atomicAdd(o + 0, hv.x * norm);
    atomicAdd(o + 1, hv.y * norm);
    atomicAdd(o + 2, hv.z * norm);
    atomicAdd(o + 3, hv.w * norm);
}

// x = (x + b)  (optionally relu), elementwise over n*128
__global__ void bias_act_kernel(float* x, const float* __restrict__ b, int n, int relu) {
    int idx = blockIdx.x * blockDim.x + threadIdx.x;
    if (idx >= n * DFEAT) return;
    float v = x[idx] + b[idx & (DFEAT - 1)];
    if (relu) v = fmaxf(v, 0.0f);
    x[idx] = v;
}

// ------------------------------- head --------------------------------------
// One 128-thread block per output row.
//   row r < 2048:            z = [x2[n_id[r]],    x2[n_id[2048+r]]] -> out[r]
//   row r >= 2048 (j=r-2048): z = [x2[n_id[j/19]], x2[n_id[4096+j]]] -> out[r]
// h = leaky_relu(z @ Wp1 + bp1, 0.2);  out = h @ Wp2 + bp2
__global__ void pred_head_kernel(const float* __restrict__ x2,
                                 const float* __restrict__ Wp1,
                                 const float* __restrict__ bp1,
                                 const float* __restrict__ Wp2,
                                 const float* __restrict__ bp2,
                                 const int* __restrict__ n_id,
                                 float* __restrict__ out) {
    __shared__ float z[2 * DFEAT];
    __shared__ float red[DFEAT];
    const int r = blockIdx.x;
    const int t = threadIdx.x;

    int ia, ib;
    if (r < NBATCH) {
        ia = n_id[r];
        ib = n_id[NBATCH + r];
    } else {
        int j = r - NBATCH;
        ia = n_id[j / NNEG];
        ib = n_id[2 * NBATCH + j];
    }
    z[t]         = x2[(long)ia * DFEAT + t];
    z[DFEAT + t] = x2[(long)ib * DFEAT + t];
    __syncthreads();

    float acc = bp1[t];
    #pragma unroll 8
    for (int k = 0; k < 2 * DFEAT; ++k) acc = fmaf(z[k], Wp1[k * DFEAT + t], acc);
    float hval = acc > 0.0f ? acc : 0.2f * acc;
    red[t] = hval * Wp2[t];
    __syncthreads();
    #pragma unroll
    for (int s = 64; s > 0; s >>= 1) {
        if (t < s) red[t] += red[t + s];
        __syncthreads();
    }
    if (t == 0) out[r] = red[0] + bp2[0];
}

// ------------------------------- launch ------------------------------------
extern "C" void kernel_launch(void* const* d_in, const int* in_sizes, int n_in,
                              void* d_out, int out_size, void* d_ws, size_t ws_size,
                              hipStream_t stream) {
    const float* id_emb = (const float*)d_in[0];
    const float* W1     = (const float*)d_in[1];
    const float* b1     = (const float*)d_in[2];
    const float* W2     = (const float*)d_in[3];
    const float* b2     = (const float*)d_in[4];
    const float* Wp1    = (const float*)d_in[5];
    const float* bp1    = (const float*)d_in[6];
    const float* Wp2    = (const float*)d_in[7];
    const float* bp2    = (const float*)d_in[8];
    const int*   eidx   = (const int*)d_in[9];     // (2, E) flat
    const int*   n_id   = (const int*)d_in[10];
    const int*   esrc   = eidx;
    const int*   edst   = eidx + NEDGES;

    // workspace layout
    char* ws = (char*)d_ws;
    float* dinv = (float*)ws;                                   // N floats
    float* buf0 = (float*)(ws + 262144);                        // N*D floats (GEMM out)
    float* buf1 = (float*)(ws + 262144 + (size_t)NNODES * DFEAT * 4); // N*D floats (agg out)
    float* logits = (float*)d_out;                              // 2048 + 38912

    const int T = 256;
    // degrees -> dinv
    deg_init_kernel<<<(NNODES + T - 1) / T, T, 0, stream>>>(dinv, NNODES);
    deg_edges_kernel<<<(NEDGES + T - 1) / T, T, 0, stream>>>(edst, dinv, NEDGES);
    dinv_kernel<<<(NNODES + T - 1) / T, T, 0, stream>>>(dinv, NNODES);

    const int mtiles = NNODES / 16;                      // 3125
    const int gemm_blocks = 1024;                        // persistent, strip-mined
    const int selfN = NNODES * (DFEAT / 4);              // 1.6M
    const int edgeN = NEDGES * (DFEAT / 4);              // 25.6M
    const int elemN = NNODES * DFEAT;                    // 6.4M

    // layer 1: h = id_emb @ W1 ; agg ; + b1 ; relu
    gemm_xw_kernel<<<gemm_blocks, 256, 0, stream>>>(id_emb, W1, buf0, mtiles);
    agg_self_kernel<<<(selfN + T - 1) / T, T, 0, stream>>>(buf0, dinv, buf1, NNODES);
    agg_edges_kernel<<<(edgeN + T - 1) / T, T, 0, stream>>>(buf0, dinv, esrc, edst, buf1, NEDGES);
    bias_act_kernel<<<(elemN + T - 1) / T, T, 0, stream>>>(buf1, b1, NNODES, 1);

    // layer 2: h = x1 @ W2 ; agg ; + b2
    gemm_xw_kernel<<<gemm_blocks, 256, 0, stream>>>(buf1, W2, buf0, mtiles);
    agg_self_kernel<<<(selfN + T - 1) / T, T, 0, stream>>>(buf0, dinv, buf1, NNODES);
    agg_edges_kernel<<<(edgeN + T - 1) / T, T, 0, stream>>>(buf0, dinv, esrc, edst, buf1, NEDGES);
    bias_act_kernel<<<(elemN + T - 1) / T, T, 0, stream>>>(buf1, b2, NNODES, 0);

    // prediction head: 2048 pos rows + 38912 neg rows
    pred_head_kernel<<<NBATCH + NBATCH * NNEG, DFEAT, 0, stream>>>(
        buf1, Wp1, bp1, Wp2, bp2, n_id, logits);
}